// MultiHeadedAttention_45646912422496
// MI455X (gfx1250) — compile-verified
//
#include <hip/hip_runtime.h>

// CDNA5 (gfx1250) wave32 WMMA, f32 path: V_WMMA_F32_16X16X4_F32
typedef __attribute__((ext_vector_type(2))) float v2f;
typedef __attribute__((ext_vector_type(8))) float v8f;

#define WMMA_F32(a, b, c) \
  __builtin_amdgcn_wmma_f32_16x16x4_f32(false, (a), false, (b), (short)0, (c), false, false)

#define SEQ 1024          // sequence length (fixed by the problem)
#define SEQ_SHIFT 10
#define SEQ_MASK 1023

// Async global->LDS (CDNA5 GLOBAL_LOAD_ASYNC_TO_LDS_B128, tracked by ASYNCcnt)
// Probe-learned signature:
//   (vsi4* src_global_generic, addrspace(3) vsi4* dst_lds, int offset, int cpol)
#if __has_builtin(__builtin_amdgcn_global_load_async_to_lds_b128)
#define USE_ASYNC_LDS 1
typedef __attribute__((__vector_size__(4 * sizeof(int)))) int vsi4;
typedef __attribute__((address_space(3))) vsi4* as3v4p;
#else
#define USE_ASYNC_LDS 0
#endif

// LDS row stride: 36 floats = 144B. Even -> K-pairs 8B aligned for b64 frag
// reads; multiple of 4 -> 16B alignment for b128 async stores; 36 = 4*9 with
// 9 odd -> the 16-lane b64 fragment read pattern is bank-conflict-free.
#define LSTR 36

// out_mode: 0 = row-major [M][N]
//           1 = head-major [B][H][SEQ][64]   (col n -> h=n/64, d=n%64)
//           2 = head-transposed [B][H][64][SEQ]
// ---------------------------------------------------------------------------
// GEMM: out = (A[MxKa] @ W[KaxN] + bias[N]) * scale
// Block tile 128M x 128N, K staged 32 at a time, double-buffered LDS with a
// single barrier per K-tile. 256 threads = 8 waves in a 4(M) x 2(N) grid;
// each wave computes 32M x 64N = 2x4 accumulators. A staged row-major
// (K contiguous), W staged column-major (K contiguous): every WMMA fragment
// is a single aligned ds_load_b64.
// ---------------------------------------------------------------------------
__global__ __launch_bounds__(256)
void gemm_bias_kernel(const float* __restrict__ A, const float* __restrict__ W,
                      const float* __restrict__ bias, float* __restrict__ out,
                      int M, int Ka, int N, float scale, int out_mode)
{
  __shared__ float As[2][128][LSTR];   // [buf][m][k]
  __shared__ float Wst[2][128][LSTR];  // [buf][n][k]  (column-major W tile)

  const int tid   = threadIdx.x;
  const int lane  = tid & 31;
  const int wave  = tid >> 5;
  const int tileN = blockIdx.x * 128;
  const int tileM = blockIdx.y * 128;

  const int mwave = wave & 3;        // 32-row M subtile
  const int nwave = wave >> 2;       // 64-col N half
  const int lm    = lane & 15;
  const int hi    = lane >> 4;       // 0/1
  const int lk    = hi * 2;          // K pair base within 4-chunk

  v8f acc[2][4] = {};

  const int nt = Ka >> 5;            // number of 32-wide K tiles
  int cur = 0;

#if !USE_ASYNC_LDS
  float4 va[4];
#endif
  float4 vw[4];

  // ---- prologue: start tile 0 ----
  #pragma unroll
  for (int r = 0; r < 4; ++r) {
    int idx = tid + r * 256;
    int row = idx >> 3;
    int c4  = (idx & 7) * 4;
#if USE_ASYNC_LDS
    __builtin_amdgcn_global_load_async_to_lds_b128(
        (vsi4*)&A[(size_t)(tileM + row) * Ka + c4],
        (as3v4p)&As[0][row][c4], 0, 0);
#else
    va[r] = *reinterpret_cast<const float4*>(&A[(size_t)(tileM + row) * Ka + c4]);
#endif
  }
  #pragma unroll
  for (int r = 0; r < 4; ++r) {
    int idx  = tid + r * 256;
    int krow = idx >> 5;
    int c4   = (idx & 31) * 4;
    vw[r] = *reinterpret_cast<const float4*>(&W[(size_t)krow * N + tileN + c4]);
  }

  for (int t = 0; t < nt; ++t) {
    // ---- commit staged tile t into LDS buffer cur ----
#if USE_ASYNC_LDS
    asm volatile("s_wait_asynccnt 0x0" ::: "memory");
#else
    #pragma unroll
    for (int r = 0; r < 4; ++r) {
      int idx = tid + r * 256;
      int row = idx >> 3;
      int c4  = (idx & 7) * 4;
      *reinterpret_cast<v2f*>(&As[cur][row][c4])     = v2f{va[r].x, va[r].y};
      *reinterpret_cast<v2f*>(&As[cur][row][c4 + 2]) = v2f{va[r].z, va[r].w};
    }
#endif
    #pragma unroll
    for (int r = 0; r < 4; ++r) {
      int idx  = tid + r * 256;
      int krow = idx >> 5;
      int c4   = (idx & 31) * 4;
      Wst[cur][c4 + 0][krow] = vw[r].x;
      Wst[cur][c4 + 1][krow] = vw[r].y;
      Wst[cur][c4 + 2][krow] = vw[r].z;
      Wst[cur][c4 + 3][krow] = vw[r].w;
    }
    __syncthreads();

    // ---- start tile t+1 (latency hidden under the 64 WMMAs below) ----
    if (t + 1 < nt) {
      int kn = (t + 1) << 5;
      #pragma unroll
      for (int r = 0; r < 4; ++r) {
        int idx = tid + r * 256;
        int row = idx >> 3;
        int c4  = (idx & 7) * 4;
#if USE_ASYNC_LDS
        __builtin_amdgcn_global_load_async_to_lds_b128(
            (vsi4*)&A[(size_t)(tileM + row) * Ka + kn + c4],
            (as3v4p)&As[cur ^ 1][row][c4], 0, 0);
#else
        va[r] = *reinterpret_cast<const float4*>(
            &A[(size_t)(tileM + row) * Ka + kn + c4]);
#endif
      }
      #pragma unroll
      for (int r = 0; r < 4; ++r) {
        int idx  = tid + r * 256;
        int krow = idx >> 5;
        int c4   = (idx & 31) * 4;
        vw[r] = *reinterpret_cast<const float4*>(
            &W[(size_t)(kn + krow) * N + tileN + c4]);
      }
    }

    // ---- compute tile t: 8 kk-steps x 8 WMMAs ----
    #pragma unroll
    for (int kk = 0; kk < 32; kk += 4) {
      v2f a0 = *reinterpret_cast<const v2f*>(&As[cur][mwave * 32 + lm][kk + lk]);
      v2f a1 = *reinterpret_cast<const v2f*>(&As[cur][mwave * 32 + 16 + lm][kk + lk]);
      #pragma unroll
      for (int tt = 0; tt < 4; ++tt) {
        v2f b = *reinterpret_cast<const v2f*>(
            &Wst[cur][nwave * 64 + tt * 16 + lm][kk + lk]);
        acc[0][tt] = WMMA_F32(a0, b, acc[0][tt]);
        acc[1][tt] = WMMA_F32(a1, b, acc[1][tt]);
      }
    }
    cur ^= 1;
  }

  // ---- epilogue: bias + scale, layout per out_mode (no integer division) ----
  #pragma unroll
  for (int ms = 0; ms < 2; ++ms) {
    #pragma unroll
    for (int tt = 0; tt < 4; ++tt) {
      int col = tileN + nwave * 64 + tt * 16 + lm;
      float bv = bias[col];
      #pragma unroll
      for (int i = 0; i < 8; ++i) {
        int row = tileM + mwave * 32 + ms * 16 + i + hi * 8;
        float v = (acc[ms][tt][i] + bv) * scale;
        size_t o;
        if (out_mode == 1) {
          int bidx = row >> SEQ_SHIFT, pos = row & SEQ_MASK;
          int h = col >> 6, d = col & 63;
          o = (((size_t)bidx * 16 + h) * SEQ + pos) * 64 + d;
        } else if (out_mode == 2) {
          int bidx = row >> SEQ_SHIFT, pos = row & SEQ_MASK;
          int h = col >> 6, d = col & 63;
          o = (((size_t)bidx * 16 + h) * 64 + d) * SEQ + pos;
        } else {
          o = (size_t)row * N + col;
        }
        out[o] = v;
      }
    }
  }
}

// ---------------------------------------------------------------------------
// Flash attention over K (SEQ), per (b,h,64-q-row tile). 4 waves/block, each
// wave owns 16 q-rows, steps 16 keys at a time. Head 0 also writes raw
// pre-mask scores (top_score). K/V slices per head = 256KB -> L2-resident.
// V is stored [B][H][64][SEQ] so every V fragment is one aligned b64 load.
// ---------------------------------------------------------------------------
__global__ __launch_bounds__(128)
void attention_kernel(const float* __restrict__ qbuf, const float* __restrict__ kbuf,
                      const float* __restrict__ vbuf, const int* __restrict__ mask,
                      const int* __restrict__ amask, float* __restrict__ ctx,
                      float* __restrict__ top)
{
  const int qt   = blockIdx.x;
  const int h    = blockIdx.y;
  const int b    = blockIdx.z;
  const int lane = threadIdx.x & 31;
  const int wave = threadIdx.x >> 5;
  const int lm   = lane & 15;
  const int hi   = lane >> 4;
  const int lk   = hi * 2;

  const size_t head_off = (((size_t)b * 16 + h) * SEQ) * 64;
  const float* qh = qbuf + head_off;   // [SEQ][64]
  const float* kh = kbuf + head_off;   // [SEQ][64]
  const float* vh = vbuf + head_off;   // [64][SEQ]  (transposed)

  const int qrow0 = qt * 64 + wave * 16;

  // q A-fragments: 16 chunks of 4 over d=64 (32 VGPRs, kept resident)
  v2f afrag[16];
  {
    const float* qr = qh + (size_t)(qrow0 + lm) * 64;
    #pragma unroll
    for (int c = 0; c < 16; ++c)
      afrag[c] = *reinterpret_cast<const v2f*>(&qr[c * 4 + lk]);
  }

  float mrow[8], lrow[8], qm[8];
  #pragma unroll
  for (int i = 0; i < 8; ++i) {
    mrow[i] = -3.0e38f;
    lrow[i] = 0.0f;
    qm[i]   = (mask[b * SEQ + qrow0 + i + hi * 8] == 0) ? 0.0f : 1.0f;
  }
  v8f oacc[4] = {};  // 16q x 64d

  __shared__ float Ps[4][16][18];    // wave-private; stride 18 -> aligned b64
  float (*P)[18] = Ps[wave];

  for (int k0 = 0; k0 < SEQ; k0 += 16) {
    // S = q @ k^T (16q x 16k), 16 WMMAs over the d dimension
    v8f s = {};
    {
      const float* kr = kh + (size_t)(k0 + lm) * 64;
      #pragma unroll
      for (int c = 0; c < 16; ++c) {
        v2f bfr = *reinterpret_cast<const v2f*>(&kr[c * 4 + lk]);
        s = WMMA_F32(afrag[c], bfr, s);
      }
    }

    // Raw pre-mask scores for head 0
    if (h == 0) {
      float* tp = top + ((size_t)b * SEQ + qrow0 + hi * 8) * SEQ + k0 + lm;
      #pragma unroll
      for (int i = 0; i < 8; ++i) tp[(size_t)i * SEQ] = s[i];
    }

    // Literal -1e18 masking (reproduces uniform softmax on masked q-rows)
    float km = (amask[b * SEQ + k0 + lm] == 0) ? 0.0f : 1.0f;
    #pragma unroll
    for (int i = 0; i < 8; ++i)
      s[i] = (qm[i] != 0.0f && km != 0.0f) ? s[i] : -1.0e18f;

    // Online softmax: row-max / row-sum across the 16 lanes of each half
    float corr[8];
    #pragma unroll
    for (int i = 0; i < 8; ++i) {
      float v = s[i];
      v = fmaxf(v, __shfl_xor(v, 1, 32));
      v = fmaxf(v, __shfl_xor(v, 2, 32));
      v = fmaxf(v, __shfl_xor(v, 4, 32));
      v = fmaxf(v, __shfl_xor(v, 8, 32));
      float newm = fmaxf(mrow[i], v);
      float p = __expf(s[i] - newm);
      s[i] = p;
      float sum = p;
      sum += __shfl_xor(sum, 1, 32);
      sum += __shfl_xor(sum, 2, 32);
      sum += __shfl_xor(sum, 4, 32);
      sum += __shfl_xor(sum, 8, 32);
      corr[i] = __expf(mrow[i] - newm);
      lrow[i] = lrow[i] * corr[i] + sum;
      mrow[i] = newm;
    }
    #pragma unroll
    for (int t = 0; t < 4; ++t)
      #pragma unroll
      for (int i = 0; i < 8; ++i)
        oacc[t][i] *= corr[i];

    // Relayout P (C-layout -> A-layout) via wave-private LDS.
    // DS ops are in-order per wave; s_wait_dscnt 0 makes the writes visible.
    #pragma unroll
    for (int i = 0; i < 8; ++i)
      P[i + hi * 8][lm] = s[i];
    asm volatile("s_wait_dscnt 0x0" ::: "memory");

    v2f pa[4];
    #pragma unroll
    for (int c = 0; c < 4; ++c)
      pa[c] = *reinterpret_cast<const v2f*>(&P[lm][c * 4 + lk]);
    asm volatile("" ::: "memory");

    // O += P @ V : M=16q, K=16keys, N=64d -> 4x4 WMMAs, V fragments are
    // single aligned global_load_b64 thanks to the [64][SEQ] V layout.
    #pragma unroll
    for (int t = 0; t < 4; ++t) {
      const float* vr = vh + (size_t)(t * 16 + lm) * SEQ + k0;
      #pragma unroll
      for (int c = 0; c < 4; ++c) {
        v2f bfr = *reinterpret_cast<const v2f*>(&vr[c * 4 + lk]);
        oacc[t] = WMMA_F32(pa[c], bfr, oacc[t]);
      }
    }
  }

  // Normalize and write ctx in [B][Q][H*64] layout (A-matrix of final GEMM)
  #pragma unroll
  for (int t = 0; t < 4; ++t) {
    #pragma unroll
    for (int i = 0; i < 8; ++i) {
      int row = qrow0 + i + hi * 8;
      ctx[((size_t)b * SEQ + row) * 1024 + h * 64 + t * 16 + lm] =
          oacc[t][i] / lrow[i];
    }
  }
}

// ---------------------------------------------------------------------------
extern "C" void kernel_launch(void* const* d_in, const int* in_sizes, int n_in,
                              void* d_out, int out_size, void* d_ws, size_t ws_size,
                              hipStream_t stream)
{
  const float* key   = (const float*)d_in[0];
  const float* value = (const float*)d_in[1];
  const float* query = (const float*)d_in[2];
  const int*   mask  = (const int*)d_in[3];
  const int*   amask = (const int*)d_in[4];
  const float* Wk    = (const float*)d_in[5];
  const float* bk    = (const float*)d_in[6];
  const float* Wv    = (const float*)d_in[7];
  const float* bv    = (const float*)d_in[8];
  const float* Wq    = (const float*)d_in[9];
  const float* bq    = (const float*)d_in[10];
  const float* Wo    = (const float*)d_in[11];
  const float* bo    = (const float*)d_in[12];

  const int B = 4, MD = 1024, MD1 = 768, HD = 1024;
  const int M = B * SEQ;  // 4096

  // Workspace: k,v,q (head layouts) + ctx = 4 x 16 MB = 64 MB (L2-resident)
  float* kbuf = (float*)d_ws;
  float* vbuf = kbuf + (size_t)M * HD;
  float* qbuf = vbuf + (size_t)M * HD;
  float* ctx  = qbuf + (size_t)M * HD;

  float* outp = (float*)d_out;                // (B, Q, 768)
  float* top  = outp + (size_t)B * SEQ * MD1; // (B, Q, K)

  dim3 blk(256);
  dim3 gproj(HD / 128, M / 128);
  // k = key@Wk + bk ; v = value@Wv + bv (transposed heads) ; q = (query@Wq + bq)/8
  gemm_bias_kernel<<<gproj, blk, 0, stream>>>(key,   Wk, bk, kbuf, M, MD,  HD, 1.0f,   1);
  gemm_bias_kernel<<<gproj, blk, 0, stream>>>(value, Wv, bv, vbuf, M, MD,  HD, 1.0f,   2);
  gemm_bias_kernel<<<gproj, blk, 0, stream>>>(query, Wq, bq, qbuf, M, MD1, HD, 0.125f, 1);

  dim3 gatt(SEQ / 64, 16, B);
  attention_kernel<<<gatt, dim3(128), 0, stream>>>(qbuf, kbuf, vbuf, mask, amask,
                                                   ctx, top);

  dim3 gout(MD1 / 128, M / 128);
  gemm_bias_kernel<<<gout, blk, 0, stream>>>(ctx, Wo, bo, outp, M, HD, MD1, 1.0f, 0);
}